// CrossAttention_69784628625684
// MI455X (gfx1250) — compile-verified
//
#include <hip/hip_runtime.h>
#include <math.h>

// ---- problem constants (match reference) ----
#define EMBED   256
#define Hh      80
#define Ww      80
#define HW      6400          // H*W
#define NHEADS  8
#define NPOINTS 4
#define Bb      8
#define MTOT    (Bb * HW)     // 51200 rows for fused-batch GEMMs

typedef float v2f __attribute__((ext_vector_type(2)));
typedef float v8f __attribute__((ext_vector_type(8)));

// ---- CDNA5 async global->LDS copy (ASYNCcnt-tracked) ----
__device__ __forceinline__ void async_copy_b128(unsigned lds_byte_addr,
                                                const void* gaddr) {
  asm volatile("global_load_async_to_lds_b128 %0, %1, off"
               :: "v"(lds_byte_addr), "v"((unsigned long long)(size_t)gaddr)
               : "memory");
}
#define S_WAIT_ASYNCCNT(n) asm volatile("s_wait_asynccnt " #n ::: "memory")

// LDS staging geometry for row-major-A GEMMs:
// chunk = 128 rows x 32 K-floats; row padded to 36 floats:
//  - async b128 stores stay 16B aligned (36*4 = 144 = 9*16)
//  - fragment ds_load_b64: lane l row-stride 36 words -> 16 distinct banks
//    (4*(9l mod 16)); K-half offset +2 words -> halves use disjoint bank sets.
#define KC      32
#define ROWPAD  36

__device__ __forceinline__ void issue_chunkA(const float* __restrict__ A,
                                             int mBlk, int kc,
                                             unsigned ldsbase, int bufi,
                                             int tid) {
  // 128 rows * 8 segs(16B) = 1024 segs; 256 threads * 4 rounds
  #pragma unroll
  for (int r = 0; r < 4; ++r) {
    int sidx = r * 256 + tid;
    int row = sidx >> 3, seg = sidx & 7;
    const float* g = A + (size_t)(mBlk + row) * EMBED + kc + seg * 4;
    unsigned lds = ldsbase +
        (unsigned)((bufi * 128 * ROWPAD + row * ROWPAD + seg * 4) * 4);
    async_copy_b128(lds, g);
  }
}

// ============================================================================
// Kernel 1: qf[b][n][c] = query[b][c][n] + sine_pe(c, n)   (LDS 32x32 transpose)
// ============================================================================
__global__ __launch_bounds__(256)
void k_prep_q(const float* __restrict__ query, float* __restrict__ qf) {
  __shared__ float tile[32][33];
  const int b  = blockIdx.z;
  const int n0 = blockIdx.x * 32;
  const int c0 = blockIdx.y * 32;
  #pragma unroll
  for (int i = 0; i < 4; ++i) {
    int c = c0 + threadIdx.y + 8 * i;
    int n = n0 + threadIdx.x;
    tile[threadIdx.y + 8 * i][threadIdx.x] =
        query[((size_t)b * EMBED + c) * HW + n];      // tile[c_local][n_local]
  }
  __syncthreads();
  #pragma unroll
  for (int i = 0; i < 4; ++i) {
    int n = n0 + threadIdx.y + 8 * i;
    int c = c0 + threadIdx.x;
    int h = n / Ww, w = n % Ww;
    int ci = c >> 2, rem = c & 3;
    float div = __expf(-(float)(2 * ci) * 0.07195578415f);  // ln(1e4)/128
    float pos = (rem < 2) ? (float)(w + 1) : (float)(h + 1);
    float ang = pos * div;
    float pe  = ((rem & 1) == 0) ? __sinf(ang) : __cosf(ang);
    qf[((size_t)b * HW + n) * EMBED + c] =
        tile[threadIdx.x][threadIdx.y + 8 * i] + pe;
  }
}

// ============================================================================
// Kernel 2: vproj = value^T @ W_value + b_value.
// A read directly in native [K][HW] layout (lane-coalesced). Wave = 16x64 tile
// (A fragment reused across 4 WMMAs). Block = 8 waves covering 128 rows.
// ============================================================================
__global__ __launch_bounds__(256)
void k_vproj_gemm(const float* __restrict__ value, const float* __restrict__ Wv,
                  const float* __restrict__ bv, float* __restrict__ vproj) {
  const int b    = blockIdx.z;
  const int wave = threadIdx.x >> 5;
  const int lane = threadIdx.x & 31;
  const int l    = lane & 15;
  const int half = lane >> 4;
  const int mBlk = blockIdx.x * 128;
  const int mw   = wave * 16;
  const int n0   = blockIdx.y * 64;

  const float* Abase = value + (size_t)b * EMBED * HW;   // [K=256][HW]
  v8f acc[4] = {};
  for (int k = 0; k < EMBED; k += 4) {
    int koff = k + 2 * half;
    v2f a;
    a.x = Abase[(size_t)koff       * HW + (mBlk + mw + l)];
    a.y = Abase[(size_t)(koff + 1) * HW + (mBlk + mw + l)];
    #pragma unroll
    for (int t = 0; t < 4; ++t) {
      v2f bm;
      bm.x = Wv[(size_t)koff       * EMBED + (n0 + t * 16 + l)];
      bm.y = Wv[(size_t)(koff + 1) * EMBED + (n0 + t * 16 + l)];
      acc[t] = __builtin_amdgcn_wmma_f32_16x16x4_f32(false, a, false, bm,
                                                     (short)0, acc[t],
                                                     false, false);
    }
  }
  float* out = vproj + ((size_t)b * HW + mBlk + mw) * EMBED + n0;
  #pragma unroll
  for (int t = 0; t < 4; ++t) {
    float bias = bv[n0 + t * 16 + l];
    #pragma unroll
    for (int r = 0; r < 8; ++r) {
      int m = r + 8 * half;
      out[(size_t)m * EMBED + t * 16 + l] = acc[t][r] + bias;
    }
  }
}

// ============================================================================
// Kernel 3: fused sampling-offset + attention-logit projection.
// One pass over qf (A) computes BOTH off = qf@W_off+b_off (N=64, tiles 0..3)
// and aw = qf@W_attn+b_attn (N=32, tiles 4..5). A is staged via double-
// buffered global_load_async_to_lds_b128 (ASYNCcnt) and read with ds_load_b64.
// ============================================================================
__global__ __launch_bounds__(256)
void k_qproj(const float* __restrict__ A,
             const float* __restrict__ W_off,  const float* __restrict__ b_off,
             const float* __restrict__ W_attn, const float* __restrict__ b_attn,
             float* __restrict__ offb, float* __restrict__ awb) {
  __shared__ float abuf[2][128 * ROWPAD];
  const unsigned ldsbase = (unsigned)(size_t)(&abuf[0][0]);
  const int wave = threadIdx.x >> 5;
  const int lane = threadIdx.x & 31;
  const int l    = lane & 15;
  const int half = lane >> 4;
  const int mBlk = blockIdx.x * 128;
  const int mw   = wave * 16;

  v8f acc[6] = {};
  issue_chunkA(A, mBlk, 0, ldsbase, 0, threadIdx.x);
  for (int c = 0; c < EMBED / KC; ++c) {
    const int cur = c & 1;
    if (c + 1 < EMBED / KC) {
      issue_chunkA(A, mBlk, (c + 1) * KC, ldsbase, 1 - cur, threadIdx.x);
      S_WAIT_ASYNCCNT(0x4);          // my 4 chunk-c ops complete (in-order)
    } else {
      S_WAIT_ASYNCCNT(0x0);
    }
    __syncthreads();                 // all waves' chunk-c data visible
    const float* ab = &abuf[cur][0];
    #pragma unroll
    for (int kk = 0; kk < KC; kk += 4) {
      v2f a = *(const v2f*)(ab + (mw + l) * ROWPAD + kk + 2 * half);
      int kg = c * KC + kk + 2 * half;
      #pragma unroll
      for (int t = 0; t < 4; ++t) {          // W_off tiles (N = 64)
        v2f bm;
        bm.x = W_off[(size_t)kg       * 64 + t * 16 + l];
        bm.y = W_off[(size_t)(kg + 1) * 64 + t * 16 + l];
        acc[t] = __builtin_amdgcn_wmma_f32_16x16x4_f32(false, a, false, bm,
                                                       (short)0, acc[t],
                                                       false, false);
      }
      #pragma unroll
      for (int t = 0; t < 2; ++t) {          // W_attn tiles (N = 32)
        v2f bm;
        bm.x = W_attn[(size_t)kg       * 32 + t * 16 + l];
        bm.y = W_attn[(size_t)(kg + 1) * 32 + t * 16 + l];
        acc[4 + t] = __builtin_amdgcn_wmma_f32_16x16x4_f32(false, a, false, bm,
                                                           (short)0, acc[4 + t],
                                                           false, false);
      }
    }
    __syncthreads();                 // protect buf[cur] before next overwrite
  }
  const size_t mrow = (size_t)(mBlk + mw);
  #pragma unroll
  for (int t = 0; t < 4; ++t) {
    float bs = b_off[t * 16 + l];
    #pragma unroll
    for (int r = 0; r < 8; ++r)
      offb[(mrow + r + 8 * half) * 64 + t * 16 + l] = acc[t][r] + bs;
  }
  #pragma unroll
  for (int t = 0; t < 2; ++t) {
    float bs = b_attn[t * 16 + l];
    #pragma unroll
    for (int r = 0; r < 8; ++r)
      awb[(mrow + r + 8 * half) * 32 + t * 16 + l] = acc[4 + t][r] + bs;
  }
}

// ============================================================================
// Kernel 4: out-projection GEMM (N=256) with fused residual, same async-LDS
// A pipeline. Wave = 16x64 tile.
// ============================================================================
__global__ __launch_bounds__(256)
void k_outproj(const float* __restrict__ A, const float* __restrict__ Wm,
               const float* __restrict__ bias, const float* __restrict__ resid,
               float* __restrict__ Cmat) {
  __shared__ float abuf[2][128 * ROWPAD];
  const unsigned ldsbase = (unsigned)(size_t)(&abuf[0][0]);
  const int wave = threadIdx.x >> 5;
  const int lane = threadIdx.x & 31;
  const int l    = lane & 15;
  const int half = lane >> 4;
  const int mBlk = blockIdx.x * 128;
  const int mw   = wave * 16;
  const int n0   = blockIdx.y * 64;

  v8f acc[4] = {};
  issue_chunkA(A, mBlk, 0, ldsbase, 0, threadIdx.x);
  for (int c = 0; c < EMBED / KC; ++c) {
    const int cur = c & 1;
    if (c + 1 < EMBED / KC) {
      issue_chunkA(A, mBlk, (c + 1) * KC, ldsbase, 1 - cur, threadIdx.x);
      S_WAIT_ASYNCCNT(0x4);
    } else {
      S_WAIT_ASYNCCNT(0x0);
    }
    __syncthreads();
    const float* ab = &abuf[cur][0];
    #pragma unroll
    for (int kk = 0; kk < KC; kk += 4) {
      v2f a = *(const v2f*)(ab + (mw + l) * ROWPAD + kk + 2 * half);
      int kg = c * KC + kk + 2 * half;
      #pragma unroll
      for (int t = 0; t < 4; ++t) {
        v2f bm;
        bm.x = Wm[(size_t)kg       * EMBED + n0 + t * 16 + l];
        bm.y = Wm[(size_t)(kg + 1) * EMBED + n0 + t * 16 + l];
        acc[t] = __builtin_amdgcn_wmma_f32_16x16x4_f32(false, a, false, bm,
                                                       (short)0, acc[t],
                                                       false, false);
      }
    }
    __syncthreads();
  }
  const size_t mrow = (size_t)(mBlk + mw);
  #pragma unroll
  for (int t = 0; t < 4; ++t) {
    float bs = bias[n0 + t * 16 + l];
    #pragma unroll
    for (int r = 0; r < 8; ++r) {
      size_t m = mrow + r + 8 * half;
      Cmat[m * EMBED + n0 + t * 16 + l] =
          acc[t][r] + bs + resid[m * EMBED + n0 + t * 16 + l];
    }
  }
}

// ============================================================================
// Kernel 5: deformable sampling. One wave = (b, q, head); lane = head-dim chan.
// ============================================================================
__global__ __launch_bounds__(256)
void k_sample(const float* __restrict__ vproj, const float* __restrict__ offb,
              const float* __restrict__ awb, float* __restrict__ attn) {
  const int hn   = threadIdx.x >> 5;
  const int lane = threadIdx.x & 31;
  const long bq  = blockIdx.x;
  const int b = (int)(bq / HW);
  const int q = (int)(bq % HW);
  const int h = q / Ww, w = q % Ww;

  float4 lg = *(const float4*)(awb + bq * (NHEADS * NPOINTS) + hn * NPOINTS);
  float mx = fmaxf(fmaxf(lg.x, lg.y), fmaxf(lg.z, lg.w));
  float e0 = __expf(lg.x - mx), e1 = __expf(lg.y - mx);
  float e2 = __expf(lg.z - mx), e3 = __expf(lg.w - mx);
  float inv = 1.f / (e0 + e1 + e2 + e3);
  float wp[4] = { e0 * inv, e1 * inv, e2 * inv, e3 * inv };

  const float* opp = offb + bq * (NHEADS * NPOINTS * 2) + hn * NPOINTS * 2;
  float4 o01 = *(const float4*)(opp);
  float4 o23 = *(const float4*)(opp + 4);
  float ox[4] = { o01.x, o01.z, o23.x, o23.z };
  float oy[4] = { o01.y, o01.w, o23.y, o23.w };

  const float* vb = vproj + (size_t)b * HW * EMBED + hn * 32 + lane;
  const float refx = (float)w * (1.f / 79.f);   // ref = (w/79, h/79)
  const float refy = (float)h * (1.f / 79.f);

  float acc = 0.f;
  #pragma unroll
  for (int p = 0; p < NPOINTS; ++p) {
    float x = refx * (float)Ww + ox[p] - 0.5f;   // loc_x*W - 0.5
    float y = refy * (float)Hh + oy[p] - 0.5f;
    float x0f = floorf(x), y0f = floorf(y);
    float wx = x - x0f, wy = y - y0f;
    int x0 = (int)x0f, y0 = (int)y0f;
    int x1 = x0 + 1,   y1 = y0 + 1;
    bool vx0 = (x0 >= 0) & (x0 < Ww), vx1 = (x1 >= 0) & (x1 < Ww);
    bool vy0 = (y0 >= 0) & (y0 < Hh), vy1 = (y1 >= 0) & (y1 < Hh);
    float g00 = (vx0 & vy0) ? vb[(size_t)(y0 * Ww + x0) * EMBED] : 0.f;
    float g10 = (vx1 & vy0) ? vb[(size_t)(y0 * Ww + x1) * EMBED] : 0.f;
    float g01 = (vx0 & vy1) ? vb[(size_t)(y1 * Ww + x0) * EMBED] : 0.f;
    float g11 = (vx1 & vy1) ? vb[(size_t)(y1 * Ww + x1) * EMBED] : 0.f;
    acc += wp[p] * (g00 * (1.f - wx) * (1.f - wy) + g10 * wx * (1.f - wy)
                  + g01 * (1.f - wx) * wy         + g11 * wx * wy);
  }
  attn[bq * EMBED + hn * 32 + lane] = acc;
}

// ============================================================================
// Kernel 6: final transpose + dual output (out+value, out)
// ============================================================================
__global__ __launch_bounds__(256)
void k_final(const float* __restrict__ T, const float* __restrict__ value,
             float* __restrict__ out) {
  __shared__ float tile[32][33];
  const int b  = blockIdx.z;
  const int n0 = blockIdx.x * 32;
  const int c0 = blockIdx.y * 32;
  #pragma unroll
  for (int i = 0; i < 4; ++i) {
    int n = n0 + threadIdx.y + 8 * i;
    int c = c0 + threadIdx.x;
    tile[threadIdx.y + 8 * i][threadIdx.x] =
        T[((size_t)b * HW + n) * EMBED + c];
  }
  __syncthreads();
  const size_t halfsz = (size_t)Bb * EMBED * HW;
  #pragma unroll
  for (int i = 0; i < 4; ++i) {
    int c = c0 + threadIdx.y + 8 * i;
    int n = n0 + threadIdx.x;
    float oi = tile[threadIdx.x][threadIdx.y + 8 * i];
    size_t idx = ((size_t)b * EMBED + c) * HW + n;
    out[idx]          = oi + value[idx];
    out[halfsz + idx] = oi;
  }
}

// ============================================================================
// Launch
// ============================================================================
extern "C" void kernel_launch(void* const* d_in, const int* in_sizes, int n_in,
                              void* d_out, int out_size, void* d_ws, size_t ws_size,
                              hipStream_t stream) {
  (void)in_sizes; (void)n_in; (void)out_size; (void)ws_size;
  const float* query   = (const float*)d_in[0];
  const float* value   = (const float*)d_in[1];
  const float* W_value = (const float*)d_in[2];
  const float* b_value = (const float*)d_in[3];
  const float* W_off   = (const float*)d_in[4];
  const float* b_off   = (const float*)d_in[5];
  const float* W_attn  = (const float*)d_in[6];
  const float* b_attn  = (const float*)d_in[7];
  const float* W_out   = (const float*)d_in[8];
  const float* b_out   = (const float*)d_in[9];
  float* out = (float*)d_out;

  // workspace (floats); vproj buffer is reused as the out-proj result T
  float* qf    = (float*)d_ws;                       // 51200*256
  float* vproj = qf    + (size_t)MTOT * EMBED;       // 51200*256
  float* offb  = vproj + (size_t)MTOT * EMBED;       // 51200*64
  float* awb   = offb  + (size_t)MTOT * 64;          // 51200*32
  float* attn  = awb   + (size_t)MTOT * 32;          // 51200*256

  // 1. qf = transpose(query) + PE
  k_prep_q<<<dim3(HW / 32, EMBED / 32, Bb), dim3(32, 8), 0, stream>>>(query, qf);

  // 2. vproj = value^T @ W_value + b_value (direct coalesced A, 16x64/wave)
  k_vproj_gemm<<<dim3(HW / 128, EMBED / 64, Bb), 256, 0, stream>>>(
      value, W_value, b_value, vproj);

  // 3. fused off/attn projections (single pass over qf, async-LDS A pipeline)
  k_qproj<<<dim3(MTOT / 128, 1, 1), 256, 0, stream>>>(
      qf, W_off, b_off, W_attn, b_attn, offb, awb);

  // 4. deformable bilinear sampling + per-point softmax
  k_sample<<<dim3(MTOT, 1, 1), 256, 0, stream>>>(vproj, offb, awb, attn);

  // 5. T = attn @ W_out + b_out + qf (residual fused; T overwrites vproj)
  k_outproj<<<dim3(MTOT / 128, EMBED / 64, 1), 256, 0, stream>>>(
      attn, W_out, b_out, qf, vproj);

  // 6. transpose back to NCHW, write both tuple outputs
  k_final<<<dim3(HW / 32, EMBED / 32, Bb), dim3(32, 8), 0, stream>>>(
      vproj, value, out);
}